// NeighborSelection_25649544691944
// MI455X (gfx1250) — compile-verified
//
#include <hip/hip_runtime.h>
#include <hip/hip_bf16.h>

#define N_NODES 100000
#define MAX_DEG 32
#define DIM     256
#define K_TOP   8

typedef __attribute__((ext_vector_type(2))) float v2f;
typedef __attribute__((ext_vector_type(8))) float v8f;

// ---------------------------------------------------------------------------
// Kernel 1: s_self / s_neigh GEMV via V_WMMA_F32_16X16X4_F32.
// One wave32 per 16 rows; block = 256 threads = 8 waves = 128 rows.
// A (16x4 f32):  lane L -> row M=L%16, K=(L<16?0:2)+v  (v = VGPR 0/1)
// B (4x16 f32):  lane L -> col N=L%16, K=(L<16?0:2)+v ; only N=0 (w_self),
//                N=1 (w_neigh) are non-zero.
// C/D (16x16):   lane L, VGPR v -> C[M = v + (L<16?0:8)][N = L%16]
// ---------------------------------------------------------------------------
__global__ __launch_bounds__(256) void ns_gemv_wmma(
    const float* __restrict__ feat,       // [N_NODES, DIM]
    const float* __restrict__ fc_w,       // [2*DIM]  (w_self | w_neigh)
    float* __restrict__ s_self,           // [N_NODES]
    float* __restrict__ s_neigh)          // [N_NODES]
{
    __shared__ float w_lds[2 * DIM];
    for (int i = threadIdx.x; i < 2 * DIM; i += 256) w_lds[i] = fc_w[i];
    __syncthreads();

    const int lane     = threadIdx.x & 31;
    const int wave     = threadIdx.x >> 5;
    const int row_base = blockIdx.x * 128 + wave * 16;
    const int m        = lane & 15;          // row (for A) / col (for B)
    const int khalf    = (lane >> 4) << 1;   // 0 or 2
    const int row      = row_base + m;
    const bool valid   = (row < N_NODES);
    const float* frow  = feat + (size_t)row * DIM;

    v8f c = {0.f, 0.f, 0.f, 0.f, 0.f, 0.f, 0.f, 0.f};

    #pragma unroll 4
    for (int k = 0; k < DIM; k += 4) {
        v2f a;
        if (valid) {
            a.x = frow[k + khalf];
            a.y = frow[k + khalf + 1];
        } else {
            a.x = 0.f; a.y = 0.f;   // keep EXEC all-1s; zero rows are harmless
        }
        v2f b;
        if (m < 2) {                 // col 0 = w_self, col 1 = w_neigh
            b.x = w_lds[m * DIM + k + khalf];
            b.y = w_lds[m * DIM + k + khalf + 1];
        } else {
            b.x = 0.f; b.y = 0.f;
        }
        // 8 args: (neg_a, A, neg_b, B, c_mod, C, reuse_a, reuse_b)
        c = __builtin_amdgcn_wmma_f32_16x16x4_f32(
                false, a, false, b, (short)0, c, false, false);
    }

    // Scatter result columns 0 (s_self) and 1 (s_neigh).
    const int mbase = (lane >> 4) << 3;   // 0 or 8
    if (m == 0) {
        #pragma unroll
        for (int v = 0; v < 8; ++v) {
            int r = row_base + mbase + v;
            if (r < N_NODES) s_self[r] = c[v];
        }
    } else if (m == 1) {
        #pragma unroll
        for (int v = 0; v < 8; ++v) {
            int r = row_base + mbase + v;
            if (r < N_NODES) s_neigh[r] = c[v];
        }
    }
}

// ---------------------------------------------------------------------------
// Kernel 2: one wave32 per node (MAX_DEG == 32 == wave width).
// Lane j owns neighbor j: gather s_neigh[nbr] (hits L2: s_neigh = 400 KB),
// score = exp(leaky_relu(...)), then 8 butterfly-argmax rounds for top-8
// with JAX tie-breaking (higher score wins; equal score -> lower position).
// ---------------------------------------------------------------------------
__global__ __launch_bounds__(256) void ns_topk(
    const float* __restrict__ s_self,
    const float* __restrict__ s_neigh,
    const long long* __restrict__ neighbors,   // [N_NODES, MAX_DEG] int64
    const float* __restrict__ fc_b,            // [1]
    float* __restrict__ out_ids,               // [N_NODES, K_TOP]
    float* __restrict__ out_scores)            // [N_NODES, K_TOP]
{
    const int lane = threadIdx.x & 31;
    const int wave = threadIdx.x >> 5;
    const int node = blockIdx.x * 8 + wave;
    if (node >= N_NODES) return;

    const float bias = fc_b[0];
    const float ss   = s_self[node];

    long long nbr = neighbors[(size_t)node * MAX_DEG + lane];
    float raw   = ss + s_neigh[nbr] + bias;
    float act   = raw > 0.f ? raw : 0.01f * raw;
    float score = expf(act);          // > 0 always
    float my_id = (float)nbr;         // ids < 2^24 -> exact in f32

    float work = score;               // working copy; consumed entries -> -1
    float res_score = 0.f, res_id = 0.f;

    #pragma unroll
    for (int kk = 0; kk < K_TOP; ++kk) {
        float bs = work;
        int   bp = lane;
        #pragma unroll
        for (int off = 16; off > 0; off >>= 1) {
            float os = __shfl_xor(bs, off, 32);
            int   op = __shfl_xor(bp, off, 32);
            if (os > bs || (os == bs && op < bp)) { bs = os; bp = op; }
        }
        // all lanes now agree on (bs, bp)
        float wid = __shfl(my_id, bp, 32);
        if (lane == kk) { res_score = bs; res_id = wid; }
        if (lane == bp) work = -1.0f;     // scores > 0, so -1 never re-wins
    }

    if (lane < K_TOP) {
        out_ids[(size_t)node * K_TOP + lane]    = res_id;
        out_scores[(size_t)node * K_TOP + lane] = res_score;
    }
}

// ---------------------------------------------------------------------------
extern "C" void kernel_launch(void* const* d_in, const int* in_sizes, int n_in,
                              void* d_out, int out_size, void* d_ws, size_t ws_size,
                              hipStream_t stream) {
    const float*     feat      = (const float*)d_in[0];       // node_features
    const float*     fc_w      = (const float*)d_in[1];       // (2*DIM,)
    const float*     fc_b      = (const float*)d_in[2];       // (1,)
    const long long* neighbors = (const long long*)d_in[3];   // int64

    float* out_ids    = (float*)d_out;                         // first output
    float* out_scores = (float*)d_out + (size_t)N_NODES * K_TOP;

    float* s_self  = (float*)d_ws;                             // 400 KB
    float* s_neigh = s_self + N_NODES;                         // 400 KB

    dim3 blk(256);
    dim3 g1((N_NODES + 127) / 128);
    ns_gemv_wmma<<<g1, blk, 0, stream>>>(feat, fc_w, s_self, s_neigh);

    dim3 g2((N_NODES + 7) / 8);
    ns_topk<<<g2, blk, 0, stream>>>(s_self, s_neigh, neighbors, fc_b,
                                    out_ids, out_scores);
}